// cdModule_27401891348709
// MI455X (gfx1250) — compile-verified
//
#include <hip/hip_runtime.h>
#include <math.h>

// Chamfer distance (squared-L2) for B=4, N=M=8192, 3-D fp32 points.
// Pairwise distances via V_WMMA_F32_16X16X4_F32 with K=4 packing:
//   A row (looped point q)  = ( qx,  qy,  qz, |q|^2 )
//   B col (owned point p)   = (-2px,-2py,-2pz, 1    )
//   => D[q,p] = |q|^2 - 2 p.q ;  dist = |p|^2 + min_q D
// Points are pre-packed as [x,y,z,|q|^2] (16B) so each lane's fragment is a
// single b64 load. Min-reduction uses raw v_min3_num_f32 (no NaN
// canonicalization). Cross-split combine via global float atomic-min
// (order-insensitive => deterministic); final fixed-order sum.

typedef float v2f __attribute__((ext_vector_type(2)));
typedef float v8f __attribute__((ext_vector_type(8)));

#define TILE 16
#define NSPLIT 8          // split the Q loop 8 ways for occupancy
#define WAVES_PER_BLOCK 8
#define BLOCK_THREADS (WAVES_PER_BLOCK * 32)

__device__ __forceinline__ float fmin3_raw(float c, float a, float b) {
    float r;
    asm("v_min3_num_f32 %0, %1, %2, %3" : "=v"(r) : "v"(c), "v"(a), "v"(b));
    return r;
}

__global__ void cd_init_inf_kernel(float* __restrict__ p, int n) {
    int i = blockIdx.x * blockDim.x + threadIdx.x;
    if (i < n) p[i] = __builtin_inff();
}

// [n,3] -> [n,4] packed (x, y, z, |q|^2)
__global__ void cd_pack_kernel(const float* __restrict__ src, float* __restrict__ dst, int n) {
    int i = blockIdx.x * blockDim.x + threadIdx.x;
    if (i < n) {
        float x = src[3 * i], y = src[3 * i + 1], z = src[3 * i + 2];
        float4 v;
        v.x = x; v.y = y; v.z = z;
        v.w = fmaf(x, x, fmaf(y, y, z * z));
        ((float4*)dst)[i] = v;
    }
}

// One wave owns 16 packed points of P (columns) and scans a 1/NSPLIT chunk of
// packed Q (rows) in 16-row tiles.
__global__ __launch_bounds__(BLOCK_THREADS)
void cd_chamfer_min_kernel(const float* __restrict__ packP, const float* __restrict__ packQ,
                           int NP, int NQ, int nBatch, float* __restrict__ distP) {
    const int lane      = threadIdx.x & 31;
    const int waveInBlk = threadIdx.x >> 5;
    const int gwave     = blockIdx.x * WAVES_PER_BLOCK + waveInBlk;

    const int tilesPerBatch = NP / TILE;
    const int wavesPerBatch = tilesPerBatch * NSPLIT;
    const int b = gwave / wavesPerBatch;
    if (b >= nBatch) return;
    const int rem   = gwave % wavesPerBatch;
    const int tile  = rem / NSPLIT;
    const int split = rem % NSPLIT;

    const v2f* Pv = (const v2f*)packP + (size_t)b * NP * 2;
    const v2f* Qv = (const v2f*)packQ + (size_t)b * NQ * 2;

    const int col = lane & 15;
    const int hi  = lane >> 4;   // 0: K0/K1 half, 1: K2/K3 half

    // Fixed 4x16 B-fragment from owned points: lanes 0-15 hold (-2px,-2py),
    // lanes 16-31 hold (-2pz, 1).
    v2f pv = Pv[(size_t)(tile * TILE + col) * 2 + hi];
    v2f bfrag;
    bfrag.x = -2.0f * pv.x;
    bfrag.y = hi ? 1.0f : (-2.0f * pv.y);
    // |p|^2 lives in the hi half's pv.y; broadcast to the lo half.
    float p2 = __shfl_xor(pv.y, 16, 32);

    const int chunk = NQ / NSPLIT;          // assumed multiple of 4*TILE
    const int q0 = split * chunk;
    const int q1 = q0 + chunk;
    const size_t laneOff = (size_t)col * 2 + hi;   // v2f index within a tile

    v8f acc0, acc1;
#pragma unroll
    for (int i = 0; i < 8; ++i) { acc0[i] = __builtin_inff(); acc1[i] = __builtin_inff(); }

    for (int q = q0; q < q1; q += 4 * TILE) {
        const v2f* qt = Qv + (size_t)q * 2 + laneOff;
        __builtin_prefetch(qt + 4 * TILE * 2, 0, 3);
        v2f a0 = qt[0 * TILE * 2];
        v2f a1 = qt[1 * TILE * 2];
        v2f a2 = qt[2 * TILE * 2];
        v2f a3 = qt[3 * TILE * 2];
        v8f z = {};
        v8f d0 = __builtin_amdgcn_wmma_f32_16x16x4_f32(false, a0, false, bfrag, (short)0, z, false, false);
        v8f d1 = __builtin_amdgcn_wmma_f32_16x16x4_f32(false, a1, false, bfrag, (short)0, z, false, false);
        v8f d2 = __builtin_amdgcn_wmma_f32_16x16x4_f32(false, a2, false, bfrag, (short)0, z, false, false);
        v8f d3 = __builtin_amdgcn_wmma_f32_16x16x4_f32(false, a3, false, bfrag, (short)0, z, false, false);
#pragma unroll
        for (int i = 0; i < 8; ++i) {
            acc0[i] = fmin3_raw(acc0[i], d0[i], d1[i]);
            acc1[i] = fmin3_raw(acc1[i], d2[i], d3[i]);
        }
    }

    // Per-lane min over the 8 row-slots, then combine lane-halves.
    float m = fminf(acc0[0], acc1[0]);
#pragma unroll
    for (int i = 1; i < 8; ++i) m = fmin3_raw(m, acc0[i], acc1[i]);
    m = fminf(m, __shfl_xor(m, 16, 32));

    float dist = p2 + m;  // add |p|^2 (constant shift per column)
    if (lane < 16) {
        atomicMin(&distP[(size_t)b * NP + tile * TILE + col], dist);
    }
}

// Fixed-order, single-block reduction: out = sum(d1)*s1 + sum(d2)*s2.
__global__ __launch_bounds__(1024)
void cd_reduce_kernel(const float* __restrict__ d1, int n1, float s1,
                      const float* __restrict__ d2, int n2, float s2,
                      float* __restrict__ out) {
    __shared__ float sm[1024];
    float a1 = 0.0f, a2 = 0.0f;
    for (int i = threadIdx.x; i < n1; i += 1024) a1 += d1[i];
    for (int i = threadIdx.x; i < n2; i += 1024) a2 += d2[i];
    sm[threadIdx.x] = a1 * s1 + a2 * s2;
    __syncthreads();
    for (int ofs = 512; ofs > 0; ofs >>= 1) {
        if (threadIdx.x < ofs) sm[threadIdx.x] += sm[threadIdx.x + ofs];
        __syncthreads();
    }
    if (threadIdx.x == 0) out[0] = sm[0];
}

extern "C" void kernel_launch(void* const* d_in, const int* in_sizes, int n_in,
                              void* d_out, int out_size, void* d_ws, size_t ws_size,
                              hipStream_t stream) {
    (void)n_in; (void)out_size; (void)ws_size;
    const int B = 4;  // per reference setup_inputs()
    const int N = in_sizes[0] / (3 * B);
    const int M = in_sizes[1] / (3 * B);

    const float* a1 = (const float*)d_in[0];  // [B, N, 3]
    const float* a2 = (const float*)d_in[1];  // [B, M, 3]
    float* out = (float*)d_out;

    // Workspace layout: dist1 [B*N] | dist2 [B*M] | pack1 [B*N*4] | pack2 [B*M*4]
    float* dist1 = (float*)d_ws;
    float* dist2 = dist1 + (size_t)B * N;
    float* pack1 = dist2 + (size_t)B * M;
    float* pack2 = pack1 + (size_t)B * N * 4;

    const int totalDist = B * (N + M);
    cd_init_inf_kernel<<<(totalDist + 255) / 256, 256, 0, stream>>>(dist1, totalDist);
    cd_pack_kernel<<<(B * N + 255) / 256, 256, 0, stream>>>(a1, pack1, B * N);
    cd_pack_kernel<<<(B * M + 255) / 256, 256, 0, stream>>>(a2, pack2, B * M);

    const int waves1  = B * (N / TILE) * NSPLIT;
    const int blocks1 = (waves1 + WAVES_PER_BLOCK - 1) / WAVES_PER_BLOCK;
    cd_chamfer_min_kernel<<<blocks1, BLOCK_THREADS, 0, stream>>>(pack1, pack2, N, M, B, dist1);

    const int waves2  = B * (M / TILE) * NSPLIT;
    const int blocks2 = (waves2 + WAVES_PER_BLOCK - 1) / WAVES_PER_BLOCK;
    cd_chamfer_min_kernel<<<blocks2, BLOCK_THREADS, 0, stream>>>(pack2, pack1, M, N, B, dist2);

    cd_reduce_kernel<<<1, 1024, 0, stream>>>(dist1, B * N, 1.0f / (B * (float)N),
                                             dist2, B * M, 1.0f / (B * (float)M), out);
}